// AQYModel_4973572129060
// MI455X (gfx1250) — compile-verified
//
#include <hip/hip_runtime.h>

// GRU recommender forward for MI455X (gfx1250, wave32).
// D(gates x batch) = W_hh(48x16) x h^T(16x16) via chained V_WMMA_F32_16X16X4_F32.
// A operand (W_hh) is loop-invariant; C operand preloaded with the 3-entry
// gx lookup table (vocab=3) + biases; gate math is elementwise in D layout.

typedef __attribute__((ext_vector_type(2))) float v2f;
typedef __attribute__((ext_vector_type(8))) float v8f;

#define HID 16
#define TSTEPS 64
#define BPB 128      // batch rows per block (8 waves x 16)
#define THREADS 256

__device__ __forceinline__ float fast_sigmoid(float x) {
  // 1 / (1 + exp(-x))  ->  v_exp_f32 + v_rcp_f32
  float e = __builtin_amdgcn_exp2f(-x * 1.44269504088896340736f);
  return __builtin_amdgcn_rcpf(1.0f + e);
}
__device__ __forceinline__ float fast_tanh(float x) {
  // 1 - 2/(exp(2x)+1)
  float e = __builtin_amdgcn_exp2f(x * 2.88539008177792681472f);
  return 1.0f - 2.0f * __builtin_amdgcn_rcpf(e + 1.0f);
}
// exchange value with lane ^ 16 (SWAPX16 ds_swizzle: xor=0x10, and=0x1f)
__device__ __forceinline__ float swap16(float x) {
  return __int_as_float(__builtin_amdgcn_ds_swizzle(__float_as_int(x), 0x401F));
}

__global__ __launch_bounds__(THREADS)
void gru_wmma_kernel(const int* __restrict__ user_id,
                     const int* __restrict__ launch_seq,
                     const float* __restrict__ user_emb,
                     const float* __restrict__ launch_emb,
                     const float* __restrict__ W_ih,
                     const float* __restrict__ W_hh,
                     const float* __restrict__ b_ih,
                     const float* __restrict__ b_hh,
                     const float* __restrict__ fc_W,
                     const float* __restrict__ fc_b,
                     float* __restrict__ out)
{
  __shared__ int   seq_s[BPB * TSTEPS];   // transposed: [t][b_local]
  __shared__ float gx_s[3 * 48];          // gx[vocab][gate]; +b_ih; +b_hh for g<32

  const int tid      = threadIdx.x;
  const int block_b0 = blockIdx.x * BPB;

  // ---- stage launch_seq into LDS, transposed (coalesced global reads) ----
  #pragma unroll
  for (int it = 0; it < (BPB * TSTEPS) / THREADS; ++it) {
    int f = it * THREADS + tid;                 // flat [b_local][t]
    int v = launch_seq[block_b0 * TSTEPS + f];
    seq_s[(f & (TSTEPS - 1)) * BPB + (f >> 6)] = v;
  }
  // ---- gx table: 3 vocab x 48 gates (tiny: 16 MACs each) ----
  if (tid < 3 * 48) {
    int vcb = tid / 48, g = tid % 48;
    float acc = b_ih[g] + (g < 32 ? b_hh[g] : 0.0f);
    #pragma unroll
    for (int k = 0; k < HID; ++k)
      acc += W_ih[g * HID + k] * launch_emb[vcb * HID + k];
    gx_s[vcb * 48 + g] = acc;
  }
  __syncthreads();

  const int lane = tid & 31;
  const int wave = tid >> 5;
  const int bl   = lane & 15;     // batch column within wave tile
  const int hi   = lane >> 4;     // lane-half (register-half selector)
  const int wb   = wave * 16 + bl;
  const int b    = block_b0 + wb;

  // ---- loop-invariant A operands: W_hh as 3 gate-tiles x 4 K-blocks ----
  // A layout (f32 16x4): lane m = l%16, VGPR v holds K = 4*kb + v + 2*hi
  v2f a[3][4];
  #pragma unroll
  for (int t = 0; t < 3; ++t) {
    #pragma unroll
    for (int kb = 0; kb < 4; ++kb) {
      const float* p = &W_hh[(t * 16 + bl) * HID + kb * 4 + hi * 2];
      a[t][kb][0] = p[0];
      a[t][kb][1] = p[1];
    }
  }
  // ---- C operand for the n-tile: b_hh only (r*gh_n needs gx kept outside) ----
  v8f cn;
  #pragma unroll
  for (int v = 0; v < 8; ++v) cn[v] = b_hh[32 + hi * 8 + v];

  float h[8], hsum[8];
  #pragma unroll
  for (int v = 0; v < 8; ++v) { h[v] = 0.0f; hsum[v] = 0.0f; }

  #pragma unroll 1
  for (int t = 0; t < TSTEPS; ++t) {
    int s = seq_s[t * BPB + wb];                    // vocab id for (b, t)
    const float* gbase = &gx_s[s * 48 + hi * 8];    // C/D layout: j = v + 8*hi
    v8f cr, cz; float gn[8];
    #pragma unroll
    for (int v = 0; v < 8; ++v) {
      cr[v] = gbase[v];        // r tile: gx + b_ih + b_hh
      cz[v] = gbase[16 + v];   // z tile: gx + b_ih + b_hh
      gn[v] = gbase[32 + v];   // n tile gx (added after r*gh_n)
    }

    // ---- relayout h (D layout j=v+8hi) -> B operands (K=4kb+v'+2hi) ----
    float q0 = swap16(hi ? h[0] : h[2]);
    float q1 = swap16(hi ? h[1] : h[3]);
    float q2 = swap16(hi ? h[4] : h[6]);
    float q3 = swap16(hi ? h[5] : h[7]);
    v2f b0, b1, b2, b3;
    b0[0] = hi ? q0   : h[0];  b0[1] = hi ? q1   : h[1];
    b1[0] = hi ? q2   : h[4];  b1[1] = hi ? q3   : h[5];
    b2[0] = hi ? h[2] : q0;    b2[1] = hi ? h[3] : q1;
    b3[0] = hi ? h[6] : q2;    b3[1] = hi ? h[7] : q3;

    // ---- gh = W_hh x h^T : 3 tiles x 4 chained K=4 f32 WMMAs ----
    v8f ar = cr, az = cz, an = cn;
    ar = __builtin_amdgcn_wmma_f32_16x16x4_f32(false, a[0][0], false, b0, (short)0, ar, false, false);
    az = __builtin_amdgcn_wmma_f32_16x16x4_f32(false, a[1][0], false, b0, (short)0, az, false, false);
    an = __builtin_amdgcn_wmma_f32_16x16x4_f32(false, a[2][0], false, b0, (short)0, an, false, false);
    ar = __builtin_amdgcn_wmma_f32_16x16x4_f32(false, a[0][1], false, b1, (short)0, ar, false, false);
    az = __builtin_amdgcn_wmma_f32_16x16x4_f32(false, a[1][1], false, b1, (short)0, az, false, false);
    an = __builtin_amdgcn_wmma_f32_16x16x4_f32(false, a[2][1], false, b1, (short)0, an, false, false);
    ar = __builtin_amdgcn_wmma_f32_16x16x4_f32(false, a[0][2], false, b2, (short)0, ar, false, false);
    az = __builtin_amdgcn_wmma_f32_16x16x4_f32(false, a[1][2], false, b2, (short)0, az, false, false);
    an = __builtin_amdgcn_wmma_f32_16x16x4_f32(false, a[2][2], false, b2, (short)0, an, false, false);
    ar = __builtin_amdgcn_wmma_f32_16x16x4_f32(false, a[0][3], false, b3, (short)0, ar, false, false);
    az = __builtin_amdgcn_wmma_f32_16x16x4_f32(false, a[1][3], false, b3, (short)0, az, false, false);
    an = __builtin_amdgcn_wmma_f32_16x16x4_f32(false, a[2][3], false, b3, (short)0, an, false, false);

    // ---- gates (elementwise in D layout, no data movement) ----
    #pragma unroll
    for (int v = 0; v < 8; ++v) {
      float r  = fast_sigmoid(ar[v]);
      float z  = fast_sigmoid(az[v]);
      float n  = fast_tanh(gn[v] + r * an[v]);
      float hn = (1.0f - z) * n + z * h[v];
      h[v] = hn;
      hsum[v] += hn;
    }
  }

  // ---- epilogue: out[b] = fc_W[:16].u + fc_W[16:].mean_t(h) + fc_b ----
  float pp = 0.0f;
  #pragma unroll
  for (int v = 0; v < 8; ++v) pp += fc_W[16 + hi * 8 + v] * hsum[v];
  pp *= (1.0f / (float)TSTEPS);

  int uid = user_id[b];
  const float* ue = &user_emb[uid * HID + hi * 8];
  float up = 0.0f;
  #pragma unroll
  for (int k = 0; k < 8; ++k) up += fc_W[hi * 8 + k] * ue[k];

  float tot = pp + up;
  tot += swap16(tot);              // combine the two lane halves
  if (hi == 0) out[b] = tot + fc_b[0];
}

extern "C" void kernel_launch(void* const* d_in, const int* in_sizes, int n_in,
                              void* d_out, int out_size, void* d_ws, size_t ws_size,
                              hipStream_t stream) {
  (void)n_in; (void)out_size; (void)d_ws; (void)ws_size;
  const int*   user_id    = (const int*)  d_in[0];
  const int*   launch_seq = (const int*)  d_in[1];
  const float* user_emb   = (const float*)d_in[2];
  const float* launch_emb = (const float*)d_in[3];
  const float* W_ih       = (const float*)d_in[4];
  const float* W_hh       = (const float*)d_in[5];
  const float* b_ih       = (const float*)d_in[6];
  const float* b_hh       = (const float*)d_in[7];
  const float* fc_W       = (const float*)d_in[8];
  const float* fc_b       = (const float*)d_in[9];

  const int B = in_sizes[0];                 // 32768
  dim3 grid(B / BPB), block(THREADS);
  gru_wmma_kernel<<<grid, block, 0, stream>>>(user_id, launch_seq, user_emb,
                                              launch_emb, W_ih, W_hh, b_ih, b_hh,
                                              fc_W, fc_b, (float*)d_out);
}